// MultiheadAttention_11819749998826
// MI455X (gfx1250) — compile-verified
//
#include <hip/hip_runtime.h>

typedef __attribute__((ext_vector_type(2))) float v2f;
typedef __attribute__((ext_vector_type(8))) float v8f;

#define BB_ 2
#define SS_ 2048
#define DD_ 1024
#define HH_ 16
#define DK_ 64

static __device__ __forceinline__ v8f wmma_f32(v2f a, v2f b, v8f c) {
  // V_WMMA_F32_16X16X4_F32 : D = A(16x4 f32) * B(4x16 f32) + C(16x16 f32)
  return __builtin_amdgcn_wmma_f32_16x16x4_f32(
      false, a, false, b, (short)0, c, false, false);
}

// ---------------------------------------------------------------------------
// GEMM: C[M=4096, N=1024] = X @ W + bias. Each wave computes one 16x16 tile.
// SPLIT=true scatters into [B,H,S,DK] head layout; else row-major [M,N].
// ---------------------------------------------------------------------------
template <bool SPLIT>
__global__ __launch_bounds__(256) void proj_kernel(
    const float* __restrict__ X, const float* __restrict__ W,
    const float* __restrict__ bias, float* __restrict__ out) {
  const int lane = threadIdx.x & 31;
  const int wid  = threadIdx.x >> 5;
  const int l15  = lane & 15;
  const int kg   = lane >> 4;           // 0/1 selects K pair within x4 step
  const int wgid = blockIdx.x * 8 + wid;
  const int tn = wgid & 63;             // 64 n-tiles
  const int tm = wgid >> 6;             // 256 m-tiles
  const int m0 = tm * 16, n0 = tn * 16;

  const float* xrow = X + (size_t)(m0 + l15) * DD_ + 2 * kg;
  const float* wcol = W + n0 + l15;

  v8f c = {};
#pragma unroll 4
  for (int k0 = 0; k0 < DD_; k0 += 4) {
    const int kk = k0 + 2 * kg;
    float2 xv = *(const float2*)(xrow + k0);
    v2f a; a.x = xv.x; a.y = xv.y;
    v2f b; b.x = wcol[(size_t)kk * DD_]; b.y = wcol[(size_t)(kk + 1) * DD_];
    c = wmma_f32(a, b, c);
  }

  const int n = n0 + l15;
  const float bv = bias[n];
#pragma unroll
  for (int r = 0; r < 8; ++r) {
    const int m = m0 + r + 8 * kg;
    const float val = c[r] + bv;
    if (SPLIT) {
      const int b_ = m >> 11, s = m & (SS_ - 1);
      const int h = n >> 6, dk = n & (DK_ - 1);
      out[(((size_t)(b_ * HH_ + h)) * SS_ + s) * DK_ + dk] = val;
    } else {
      out[(size_t)m * DD_ + n] = val;
    }
  }
}

// ---------------------------------------------------------------------------
// Attention: one block per (b,h, 16-row q tile). Scores+P live in LDS.
// ---------------------------------------------------------------------------
#define SCSTRIDE 2052  // 2048 + 4 pad (keeps 8B alignment, spreads banks)

__global__ __launch_bounds__(256) void attn_kernel(
    const float* __restrict__ qh, const float* __restrict__ kh,
    const float* __restrict__ vh, const unsigned char* __restrict__ mask,
    const float* __restrict__ bias, float* __restrict__ p_out,
    float* __restrict__ attn_out) {
  __shared__ float sc[16 * SCSTRIDE];   // 131328 B : scores -> probabilities
  __shared__ float part[4 * 32 * 8];    //   4096 B : PV partial combine

  const int lane = threadIdx.x & 31;
  const int wid  = threadIdx.x >> 5;
  const int l15  = lane & 15;
  const int kg   = lane >> 4;
  const int bh = blockIdx.x >> 7;           // b*H + h
  const int q0 = (blockIdx.x & 127) << 4;   // q-tile base row
  const int bb = bh >> 4;                   // batch
  const int hh = bh & 15;                   // head

  // ---------------- phase 1: scores = Q K^T / sqrt(DK), masked ----------------
  v2f qa[16];
  {
    const float* qptr = qh + ((size_t)bh * SS_ + q0 + l15) * DK_ + 2 * kg;
#pragma unroll
    for (int t = 0; t < 16; ++t) {
      float2 qv = *(const float2*)(qptr + 4 * t);
      qa[t].x = qv.x; qa[t].y = qv.y;
    }
  }
  for (int kt = 0; kt < 16; ++kt) {
    const int kbase = wid * 256 + kt * 16;
    const float* kptr = kh + ((size_t)bh * SS_ + kbase + l15) * DK_ + 2 * kg;
    v8f c = {};
#pragma unroll
    for (int t = 0; t < 16; ++t) {
      float2 kv = *(const float2*)(kptr + 4 * t);
      v2f b; b.x = kv.x; b.y = kv.y;
      c = wmma_f32(qa[t], b, c);
    }
    const int key = kbase + l15;
    const bool mk = mask[(size_t)bb * SS_ + key] != 0;
#pragma unroll
    for (int r = 0; r < 8; ++r)
      sc[(r + 8 * kg) * SCSTRIDE + key] = mk ? -__builtin_inff()
                                             : c[r] * 0.125f;
  }
  __syncthreads();

  // -------- phase 2: softmax, * attn_bias, emit p_attn, keep P in LDS --------
  for (int rr = 0; rr < 2; ++rr) {
    const int qr = wid * 2 + rr;
    float* srow = &sc[qr * SCSTRIDE];
    float mx = -__builtin_inff();
    for (int k = lane; k < SS_; k += 32) mx = fmaxf(mx, srow[k]);
#pragma unroll
    for (int off = 16; off; off >>= 1) mx = fmaxf(mx, __shfl_xor(mx, off, 32));
    float sum = 0.f;
    for (int k = lane; k < SS_; k += 32) {
      float e = __expf(srow[k] - mx);
      srow[k] = e;
      sum += e;
    }
#pragma unroll
    for (int off = 16; off; off >>= 1) sum += __shfl_xor(sum, off, 32);
    const float inv = 1.0f / sum;
    const float* brow = bias + ((size_t)bh * SS_ + q0 + qr) * SS_;
    float* prow = p_out + ((size_t)bh * SS_ + q0 + qr) * SS_;
    for (int k = lane; k < SS_; k += 32) {
      const float p = srow[k] * inv * brow[k];
      srow[k] = p;
      prow[k] = p;
    }
  }
  __syncthreads();

  // ---------------- phase 3: out_tile = P @ V  (16 x 64) ----------------
  const int dt    = wid & 3;   // d-column tile (16 wide)
  const int khalf = wid >> 2;  // key half (1024 keys each)
  const float* vptr = vh + (size_t)bh * SS_ * DK_ + dt * 16 + l15;
  const float* arow = &sc[l15 * SCSTRIDE];
  v8f c = {};
  const int kend = khalf * 1024 + 1024;
#pragma unroll 4
  for (int k0 = khalf * 1024; k0 < kend; k0 += 4) {
    const int kk = k0 + 2 * kg;
    float2 av = *(const float2*)(arow + kk);      // ds_load_b64 from LDS
    v2f a; a.x = av.x; a.y = av.y;
    v2f b; b.x = vptr[(size_t)kk * DK_]; b.y = vptr[(size_t)(kk + 1) * DK_];
    c = wmma_f32(a, b, c);
  }
  if (khalf == 1) {
    float* pp = &part[(dt * 32 + lane) * 8];
#pragma unroll
    for (int r = 0; r < 8; ++r) pp[r] = c[r];
  }
  __syncthreads();
  if (khalf == 0) {
    const float* pp = &part[(dt * 32 + lane) * 8];
#pragma unroll
    for (int r = 0; r < 8; ++r) {
      const float tot = c[r] + pp[r];
      const int mq = r + 8 * kg;
      attn_out[((size_t)bb * SS_ + q0 + mq) * DD_ + hh * DK_ + dt * 16 + l15] =
          tot;
    }
  }
}

// ---------------------------------------------------------------------------
extern "C" void kernel_launch(void* const* d_in, const int* in_sizes, int n_in,
                              void* d_out, int out_size, void* d_ws,
                              size_t ws_size, hipStream_t stream) {
  const float* q  = (const float*)d_in[0];
  const float* k  = (const float*)d_in[1];
  const float* v  = (const float*)d_in[2];
  const unsigned char* mask = (const unsigned char*)d_in[3];
  const float* attn_bias = (const float*)d_in[4];
  const float* Wq = (const float*)d_in[5];
  const float* bq = (const float*)d_in[6];
  const float* Wk = (const float*)d_in[7];
  const float* bk = (const float*)d_in[8];
  const float* Wv = (const float*)d_in[9];
  const float* bv = (const float*)d_in[10];
  const float* Wo = (const float*)d_in[11];
  const float* bo = (const float*)d_in[12];

  float* ws = (float*)d_ws;
  const size_t SEG = (size_t)BB_ * SS_ * DD_;  // 4,194,304 floats
  float* qh = ws;
  float* kh = ws + SEG;
  float* vh = ws + 2 * SEG;
  float* ao = ws + 3 * SEG;

  float* out   = (float*)d_out;        // [B,S,D]
  float* p_out = out + SEG;            // [B,H,S,S]

  dim3 blk(256);
  proj_kernel<true><<<2048, blk, 0, stream>>>(q, Wq, bq, qh);
  proj_kernel<true><<<2048, blk, 0, stream>>>(k, Wk, bk, kh);
  proj_kernel<true><<<2048, blk, 0, stream>>>(v, Wv, bv, vh);
  attn_kernel<<<4096, blk, 0, stream>>>(qh, kh, vh, mask, attn_bias, p_out, ao);
  proj_kernel<false><<<2048, blk, 0, stream>>>(ao, Wo, bo, out);
}